// CapsuleNet_35218731828078
// MI455X (gfx1250) — compile-verified
//
#include <hip/hip_runtime.h>
#include <math.h>

// Problem constants (match reference)
#define NN     10000
#define MM     32
#define NFEATC 512
#define NCAPS  8
#define NHIDD  16
#define NCLS   16
#define REPD   128   // NCAPS*NHID
#define ROUTIT 6
#define KNNK   8

typedef float f32x2 __attribute__((ext_vector_type(2)));
typedef float f32x8 __attribute__((ext_vector_type(8)));

// V_WMMA_F32_16X16X4_F32 : D(16x16 f32) = A(16x4 f32) * B(4x16 f32) + C
// Lane layout (wave32): A: lanes0-15 hold row M=lane {K=k,k+1}, lanes16-31 row M=lane-16 {K=k+2,k+3}
//                       B: lanes0-15 col N=lane {K=k,k+1},     lanes16-31 col N=lane-16 {K=k+2,k+3}
//                       C/D: VGPR r, lanes0-15 -> (M=r, N=lane); lanes16-31 -> (M=r+8, N=lane-16)
static __device__ __forceinline__ f32x8 wmma4(f32x2 a, f32x2 b, f32x8 c) {
  return __builtin_amdgcn_wmma_f32_16x16x4_f32(false, a, false, b, (short)0, c, false, false);
}

// ---------------------------------------------------------------------------
// Kernel 1: h = relu(x @ pca_W + pca_b)   [10000x512]x[512x128]
// grid 625 (row tiles), block 256 = 8 waves, wave w -> col tile w (8*16=128)
// ---------------------------------------------------------------------------
__global__ __launch_bounds__(256) void k_gemm_pca(
    const float* __restrict__ A, const float* __restrict__ W,
    const float* __restrict__ bias, float* __restrict__ out) {
  const int lane = threadIdx.x & 31, wave = threadIdx.x >> 5;
  const int row0 = blockIdx.x * 16, col0 = wave * 16;
  const int m = lane & 15, hi = lane >> 4;
  f32x8 c = {};
  for (int k = 0; k < NFEATC; k += 4) {
    const int koff = k + (hi << 1);
    f32x2 a, b;
    a.x = A[(row0 + m) * NFEATC + koff];
    a.y = A[(row0 + m) * NFEATC + koff + 1];
    b.x = W[koff * REPD + col0 + m];
    b.y = W[(koff + 1) * REPD + col0 + m];
    c = wmma4(a, b, c);
  }
  const float bc = bias[col0 + m];
#pragma unroll
  for (int r = 0; r < 8; ++r) {
    const int row = row0 + r + hi * 8;
    out[row * REPD + col0 + m] = fmaxf(c[r] + bc, 0.f);
  }
}

// ---------------------------------------------------------------------------
// Kernel 2: per-(node,capsule) l2 normalize -> xn[10001 x 128], row 10000 = 0 pad
// ---------------------------------------------------------------------------
__global__ __launch_bounds__(256) void k_normcap(
    const float* __restrict__ h, float* __restrict__ xn) {
  const int idx = blockIdx.x * 256 + threadIdx.x;
  if (idx >= (NN + 1) * NCAPS) return;
  const int n = idx >> 3, f = idx & 7;
  float* o = xn + (size_t)n * REPD + f * NHIDD;
  if (n == NN) {
#pragma unroll
    for (int d = 0; d < NHIDD; ++d) o[d] = 0.f;
    return;
  }
  const float* src = h + (size_t)n * REPD + f * NHIDD;
  float v[NHIDD], s = 0.f;
#pragma unroll
  for (int d = 0; d < NHIDD; ++d) { v[d] = src[d]; s = fmaf(v[d], v[d], s); }
  const float inv = 1.f / fmaxf(sqrtf(s), 1e-12f);
#pragma unroll
  for (int d = 0; d < NHIDD; ++d) o[d] = v[d] * inv;
}

// ---------------------------------------------------------------------------
// Kernel 3: disentangled routing. One 128-thread block per node; the 32x128
// neighbor slab lives in LDS (16KB) and is reused across all 6 iterations.
// ---------------------------------------------------------------------------
__global__ __launch_bounds__(128) void k_routing(
    const float* __restrict__ xn, const int* __restrict__ nb,
    float* __restrict__ out, int do_relu) {
  __shared__ float z[MM][REPD];   // gathered normalized neighbors
  __shared__ float xk[REPD];
  __shared__ float u[REPD];
  __shared__ float p[MM][NCAPS];
  const int node = blockIdx.x;
  const int t = threadIdx.x;
  const int cap = t >> 4;

  xk[t] = xn[(size_t)node * REPD + t];
#pragma unroll 4
  for (int m = 0; m < MM; ++m) {
    const int j = nb[node * MM + m];          // [0, NN]; row NN is zero pad
    z[m][t] = xn[(size_t)j * REPD + t];
  }
  __syncthreads();

  for (int it = 0; it < ROUTIT; ++it) {
    if (it == 0) {  // softmax of zeros == 1/NCAPS
      float s = 0.f;
#pragma unroll 4
      for (int m = 0; m < MM; ++m) s += z[m][t];
      u[t] = s * (1.f / NCAPS) + xk[t];
    } else {
      // p[m][c] = <z[m,c,:], u[c,:]>  (256 dot-16's over 128 threads)
      for (int pr = t; pr < MM * NCAPS; pr += 128) {
        const int m = pr >> 3, c = pr & 7;
        float s = 0.f;
#pragma unroll
        for (int d = 0; d < NHIDD; ++d) s = fmaf(z[m][c * NHIDD + d], u[c * NHIDD + d], s);
        p[m][c] = s;
      }
      __syncthreads();
      if (t < MM) {  // softmax over capsules for row t
        float mx = p[t][0];
#pragma unroll
        for (int c = 1; c < NCAPS; ++c) mx = fmaxf(mx, p[t][c]);
        float e[NCAPS], s = 0.f;
#pragma unroll
        for (int c = 0; c < NCAPS; ++c) { e[c] = expf(p[t][c] - mx); s += e[c]; }
        const float inv = 1.f / s;
#pragma unroll
        for (int c = 0; c < NCAPS; ++c) p[t][c] = e[c] * inv;
      }
      __syncthreads();
      float s = 0.f;
#pragma unroll 4
      for (int m = 0; m < MM; ++m) s = fmaf(z[m][t], p[m][cap], s);
      u[t] = s + xk[t];
    }
    if (it < ROUTIT - 1) {  // l2norm per capsule
      __syncthreads();
      float nrm = 0.f;
#pragma unroll
      for (int d = 0; d < NHIDD; ++d) { const float w = u[cap * NHIDD + d]; nrm = fmaf(w, w, nrm); }
      nrm = fmaxf(sqrtf(nrm), 1e-12f);
      const float val = u[t];
      __syncthreads();
      u[t] = val / nrm;
    }
    __syncthreads();
  }
  const float v = u[t];
  out[(size_t)node * REPD + t] = do_relu ? fmaxf(v, 0.f) : v;
}

// ---------------------------------------------------------------------------
// Kernel 4: per-capsule squared norms sq[f][n]
// ---------------------------------------------------------------------------
__global__ __launch_bounds__(256) void k_sq(
    const float* __restrict__ hx, float* __restrict__ sq) {
  const int idx = blockIdx.x * 256 + threadIdx.x;
  if (idx >= NCAPS * NN) return;
  const int f = idx / NN, n = idx % NN;
  const float* src = hx + (size_t)n * REPD + f * NHIDD;
  float s = 0.f;
#pragma unroll
  for (int d = 0; d < NHIDD; ++d) s = fmaf(src[d], src[d], s);
  sq[idx] = s;
}

// ---------------------------------------------------------------------------
// Kernel 5: per-row 8th-smallest squared distance (sqrt elided: monotone).
// Wave per (cap,row); block = 8 waves shares LDS-staged X_j chunks of 256.
// Diagonal forced to 0 (matches reference's eye-zeroed D before top_k).
// grid (1250, 8), block 256.
// ---------------------------------------------------------------------------
__global__ __launch_bounds__(256) void k_thresh(
    const float* __restrict__ hx, const float* __restrict__ sq,
    float* __restrict__ th) {
  const int f = blockIdx.y;
  const int wave = threadIdx.x >> 5, lane = threadIdx.x & 31;
  const int i = blockIdx.x * 8 + wave;            // exact: 1250*8 = 10000
  __shared__ float Xs[256 * 17];                  // padded stride vs bank conflicts
  __shared__ float sqs[256];

  float xi[NHIDD];
#pragma unroll
  for (int d = 0; d < NHIDD; ++d) xi[d] = hx[(size_t)i * REPD + f * NHIDD + d];
  const float sqi = sq[f * NN + i];

  float a8[KNNK];                                  // sorted ascending, register-resident
#pragma unroll
  for (int q = 0; q < KNNK; ++q) a8[q] = 3.4e38f;

  for (int c0 = 0; c0 < NN; c0 += 256) {
    {
      const int j = c0 + threadIdx.x;
      if (j < NN) {
#pragma unroll
        for (int d = 0; d < NHIDD; ++d)
          Xs[threadIdx.x * 17 + d] = hx[(size_t)j * REPD + f * NHIDD + d];
        sqs[threadIdx.x] = sq[f * NN + j];
      } else {
#pragma unroll
        for (int d = 0; d < NHIDD; ++d) Xs[threadIdx.x * 17 + d] = 0.f;
        sqs[threadIdx.x] = 3.4e38f;               // padded j never selected
      }
    }
    __syncthreads();
    for (int jj = lane; jj < 256; jj += 32) {
      const int j = c0 + jj;
      if (j >= NN) break;
      float dot = 0.f;
#pragma unroll
      for (int d = 0; d < NHIDD; ++d) dot = fmaf(xi[d], Xs[jj * 17 + d], dot);
      float dsq = fmaxf(sqi + sqs[jj] - 2.f * dot, 0.f);
      if (j == i) dsq = 0.f;
      if (dsq < a8[KNNK - 1]) {                    // replace max, one bubble pass
        a8[KNNK - 1] = dsq;
#pragma unroll
        for (int q = KNNK - 1; q > 0; --q) {
          const float lo = fminf(a8[q - 1], a8[q]);
          const float hh = fmaxf(a8[q - 1], a8[q]);
          a8[q - 1] = lo; a8[q] = hh;
        }
      }
    }
    __syncthreads();
  }

  // Extract global k-th smallest across 32 lanes x 8 sorted lists.
  float kth = 0.f;
#pragma unroll
  for (int t = 0; t < KNNK; ++t) {
    float bv = a8[0];
    int bl = lane;
    for (int off = 16; off >= 1; off >>= 1) {
      const float ov = __shfl_xor(bv, off, 32);
      const int   ol = __shfl_xor(bl, off, 32);
      if (ov < bv || (ov == bv && ol < bl)) { bv = ov; bl = ol; }
    }
    if (bl == lane) {                              // winner pops exactly one instance
#pragma unroll
      for (int q = 0; q < KNNK - 1; ++q) a8[q] = a8[q + 1];
      a8[KNNK - 1] = 3.4e38f;
    }
    kth = bv;
  }
  if (lane == 0) th[f * NN + i] = kth;
}

// ---------------------------------------------------------------------------
// Kernel 6: KNN-GCN aggregate, WMMA throughout.
//  out_i = relu( (X_i + sum_{j!=i} w_ij X_j) / (1 + deg_i) ),
//  w_ij = 0.5*((d_ij<=t_i)+(d_ij<=t_j)),  deg_i = sum_j w_ij.
// Per wave: one 16-row i-tile x one capsule; stream j in 256-wide LDS chunks.
// Per 16x16 j-tile: 4x WMMA Gram -> w in C-layout -> LDS 16x17 -> 4x WMMA w@X.
// grid (79, 8), block 256 (8 waves -> 8 consecutive i-tiles).
// ---------------------------------------------------------------------------
__global__ __launch_bounds__(256) void k_agg(
    const float* __restrict__ hx, const float* __restrict__ sq,
    const float* __restrict__ th, float* __restrict__ hout) {
  const int f = blockIdx.y;
  const int wave = threadIdx.x >> 5, lane = threadIdx.x & 31;
  const int itile = blockIdx.x * 8 + wave;
  const int m = lane & 15, hi = lane >> 4;
  const bool active = (itile < NN / 16);
  const int i0 = itile * 16;

  __shared__ float Xs[256 * 17];
  __shared__ float sqs[256], tjs[256];
  __shared__ float wls[8][16 * 17];

  f32x2 aG[4];
  float sqi[8], ti[8];
  if (active) {
#pragma unroll
    for (int ks = 0; ks < 4; ++ks) {
      const int koff = ks * 4 + (hi << 1);
      aG[ks].x = hx[(size_t)(i0 + m) * REPD + f * NHIDD + koff];
      aG[ks].y = hx[(size_t)(i0 + m) * REPD + f * NHIDD + koff + 1];
    }
#pragma unroll
    for (int r = 0; r < 8; ++r) {
      const int row = i0 + r + hi * 8;
      sqi[r] = sq[f * NN + row];
      ti[r]  = th[f * NN + row];
    }
  }

  f32x8 acc = {};
  float deg[8] = {0.f, 0.f, 0.f, 0.f, 0.f, 0.f, 0.f, 0.f};

  for (int c0 = 0; c0 < NN; c0 += 256) {
    {
      const int j = c0 + threadIdx.x;
      if (j < NN) {
#pragma unroll
        for (int d = 0; d < NHIDD; ++d)
          Xs[threadIdx.x * 17 + d] = hx[(size_t)j * REPD + f * NHIDD + d];
        sqs[threadIdx.x] = sq[f * NN + j];
        tjs[threadIdx.x] = th[f * NN + j];
      } else {
#pragma unroll
        for (int d = 0; d < NHIDD; ++d) Xs[threadIdx.x * 17 + d] = 0.f;
        sqs[threadIdx.x] = 0.f;
        tjs[threadIdx.x] = -1.f;
      }
    }
    __syncthreads();
    if (active) {
      for (int jt = 0; jt < 16; ++jt) {
        const int jbase = c0 + jt * 16;
        if (jbase >= NN) break;                    // uniform per wave
        // Gram tile G[i,j] = <X_i, X_j>  (K = 16 -> 4 WMMA steps)
        f32x8 g = {};
#pragma unroll
        for (int ks = 0; ks < 4; ++ks) {
          const int koff = ks * 4 + (hi << 1);
          f32x2 b;
          b.x = Xs[(jt * 16 + m) * 17 + koff];
          b.y = Xs[(jt * 16 + m) * 17 + koff + 1];
          g = wmma4(aG[ks], b, g);
        }
        const float sqj = sqs[jt * 16 + m];
        const float tj  = tjs[jt * 16 + m];
        const int j = jbase + m;
#pragma unroll
        for (int r = 0; r < 8; ++r) {
          const int row = i0 + r + hi * 8;
          const float dsq = fmaxf(sqi[r] + sqj - 2.f * g[r], 0.f);
          float w = 0.f;
          if (j != row)
            w = 0.5f * ((dsq <= ti[r] ? 1.f : 0.f) + (dsq <= tj ? 1.f : 0.f));
          deg[r] += w;
          wls[wave][(r + hi * 8) * 17 + m] = w;    // wave-private, in-order LDS
        }
        // acc[i,d] += W(16x16) @ X_j(16x16)
#pragma unroll
        for (int ks = 0; ks < 4; ++ks) {
          const int koff = ks * 4 + (hi << 1);
          f32x2 a2, b2;
          a2.x = wls[wave][m * 17 + koff];
          a2.y = wls[wave][m * 17 + koff + 1];
          b2.x = Xs[(jt * 16 + koff) * 17 + m];
          b2.y = Xs[(jt * 16 + koff + 1) * 17 + m];
          acc = wmma4(a2, b2, acc);
        }
      }
    }
    __syncthreads();
  }

  if (active) {
#pragma unroll
    for (int r = 0; r < 8; ++r) {                  // row-degree: sum 16 lanes in half
      float d = deg[r];
      for (int off = 1; off < 16; off <<= 1) d += __shfl_xor(d, off, 32);
      deg[r] = d;
    }
#pragma unroll
    for (int r = 0; r < 8; ++r) {
      const int row = i0 + r + hi * 8;
      const float self = hx[(size_t)row * REPD + f * NHIDD + m];
      const float v = (acc[r] + self) / (1.f + deg[r]);
      hout[(size_t)row * REPD + f * NHIDD + m] = fmaxf(v, 0.f);
    }
  }
}

// ---------------------------------------------------------------------------
// Kernel 7: logits = h @ mlp_W + b, then row log_softmax in-register
// (row lives across 16 lanes of a half -> shuffle max/sum). grid 79, block 256.
// ---------------------------------------------------------------------------
__global__ __launch_bounds__(256) void k_mlp_lsm(
    const float* __restrict__ h, const float* __restrict__ W,
    const float* __restrict__ bias, float* __restrict__ out) {
  const int wave = threadIdx.x >> 5, lane = threadIdx.x & 31;
  const int itile = blockIdx.x * 8 + wave;
  if (itile >= NN / 16) return;                    // wave-uniform, no barriers here
  const int m = lane & 15, hi = lane >> 4;
  const int row0 = itile * 16;
  f32x8 c = {};
  for (int k = 0; k < REPD; k += 4) {
    const int koff = k + (hi << 1);
    f32x2 a, b;
    a.x = h[(size_t)(row0 + m) * REPD + koff];
    a.y = h[(size_t)(row0 + m) * REPD + koff + 1];
    b.x = W[koff * NCLS + m];
    b.y = W[(koff + 1) * NCLS + m];
    c = wmma4(a, b, c);
  }
  const float bc = bias[m];
#pragma unroll
  for (int r = 0; r < 8; ++r) {
    const float v = c[r] + bc;
    float mx = v;
    for (int off = 1; off < 16; off <<= 1) mx = fmaxf(mx, __shfl_xor(mx, off, 32));
    float s = expf(v - mx);
    for (int off = 1; off < 16; off <<= 1) s += __shfl_xor(s, off, 32);
    const int row = row0 + r + hi * 8;
    out[row * NCLS + m] = (v - mx) - logf(s);
  }
}

// ---------------------------------------------------------------------------
// Host launcher. Inputs: x, nb, pca_W, pca_b, mlp_W, mlp_b.
// d_out = [log_softmax 10000x16 | h 10000x128 | hidden_x 10000x128].
// Workspace use: ~10.9 MB of floats.
// ---------------------------------------------------------------------------
extern "C" void kernel_launch(void* const* d_in, const int* in_sizes, int n_in,
                              void* d_out, int out_size, void* d_ws, size_t ws_size,
                              hipStream_t stream) {
  const float* x     = (const float*)d_in[0];
  const int*   nb    = (const int*)  d_in[1];
  const float* pca_W = (const float*)d_in[2];
  const float* pca_b = (const float*)d_in[3];
  const float* mlp_W = (const float*)d_in[4];
  const float* mlp_b = (const float*)d_in[5];

  float* ws    = (float*)d_ws;
  float* xn    = ws;                                   // (NN+1)*REPD = 1,280,128
  float* tmpA  = ws + (size_t)(NN + 1) * REPD;         // NN*REPD     = 1,280,000
  float* sqb   = tmpA + (size_t)NN * REPD;             // NCAPS*NN    = 80,000
  float* thb   = sqb + (size_t)NCAPS * NN;             // NCAPS*NN    = 80,000

  float* out     = (float*)d_out;
  float* out_lsm = out;                                // 160,000
  float* out_h   = out + (size_t)NN * NCLS;            // 1,280,000
  float* out_hx  = out_h + (size_t)NN * REPD;          // 1,280,000

  const int normGrid = ((NN + 1) * NCAPS + 255) / 256;

  // PCA + ReLU
  k_gemm_pca<<<NN / 16, 256, 0, stream>>>(x, pca_W, pca_b, tmpA);
  // Layer 0: normalize -> route -> ReLU
  k_normcap<<<normGrid, 256, 0, stream>>>(tmpA, xn);
  k_routing<<<NN, 128, 0, stream>>>(xn, nb, tmpA, 1);
  // Layer 1: normalize -> route (raw output == hidden_x)
  k_normcap<<<normGrid, 256, 0, stream>>>(tmpA, xn);
  k_routing<<<NN, 128, 0, stream>>>(xn, nb, out_hx, 0);
  // KNN graph + GCN aggregate per capsule, then ReLU -> h
  k_sq<<<(NCAPS * NN + 255) / 256, 256, 0, stream>>>(out_hx, sqb);
  k_thresh<<<dim3(NN / 8, NCAPS), 256, 0, stream>>>(out_hx, sqb, thb);
  k_agg<<<dim3((NN / 16 + 7) / 8, NCAPS), 256, 0, stream>>>(out_hx, sqb, thb, out_h);
  // Classifier + log_softmax
  k_mlp_lsm<<<(NN / 16 + 7) / 8, 256, 0, stream>>>(out_h, mlp_W, mlp_b, out_lsm);
}